// StackRNNCell_37357625540657
// MI455X (gfx1250) — compile-verified
//
#include <hip/hip_runtime.h>

typedef __attribute__((ext_vector_type(2))) float v2f;
typedef __attribute__((ext_vector_type(8))) float v8f;

static constexpr int kB  = 8192;
static constexpr int kD  = 64;
static constexpr int kW  = 128;
static constexpr int kH  = 128;
static constexpr int kKF = 384;   // concat width
static constexpr int FSTR = 388;  // padded LDS stride for flat (388 % 64 == 4 -> conflict free)
static constexpr int ASTR = 132;  // padded LDS stride for A tiles (132 % 64 == 4)

__device__ __forceinline__ v8f wmma4(v2f a, v2f b, v8f c) {
  // D = A(16x4,f32) * B(4x16,f32) + C(16x16,f32); exact f32 path on CDNA5
  return __builtin_amdgcn_wmma_f32_16x16x4_f32(false, a, false, b, (short)0, c,
                                               false, false);
}

__global__ __launch_bounds__(256)
void srnn_gemm_kernel(const float* __restrict__ stack,
                      const float* __restrict__ state_prev,
                      const float* __restrict__ x_emb,
                      const float* __restrict__ W_state,
                      const float* __restrict__ b_state,
                      const float* __restrict__ W_top,
                      const float* __restrict__ b_top,
                      const float* __restrict__ W_mem,
                      const float* __restrict__ b_mem,
                      const float* __restrict__ W_buf,
                      const float* __restrict__ b_buf,
                      const float* __restrict__ W_so,
                      const float* __restrict__ b_so,
                      float* __restrict__ out_state,
                      float* __restrict__ out_buf,
                      float* __restrict__ out_probs) {
  __shared__ float flat[16][FSTR];  // [x_emb | state_emb | top_emb] per row
  __shared__ float Abuf[16][ASTR];  // staging for WMMA A operand tiles
  __shared__ float lm[16][3];

  const int tid  = threadIdx.x;
  const int wv   = tid >> 5;       // wave id, 0..7 (wave32)
  const int lane = tid & 31;
  const int l15  = lane & 15;      // A: row M; B/D: col N
  const int kh   = lane >> 4;      // K-half selector for A/B
  const int n0   = wv * 16;        // this wave's N tile
  const int m0   = blockIdx.x * 16;

  // ---- load x_emb tile -> flat[:,0:128]; state_prev tile -> Abuf (float4, coalesced)
  {
    const float4* xs = (const float4*)(x_emb + (size_t)m0 * kH);
    const float4* ss = (const float4*)(state_prev + (size_t)m0 * kH);
    for (int i = tid; i < 16 * 32; i += 256) {
      int r = i >> 5, c4 = i & 31;
      *(float4*)&flat[r][c4 * 4] = xs[r * 32 + c4];
      *(float4*)&Abuf[r][c4 * 4] = ss[r * 32 + c4];
    }
  }
  __syncthreads();

  // ---- stage A1: state_emb = state_prev @ W_state + b_state  (K = 128)
  {
    v8f acc = {};
#pragma unroll 8
    for (int k0 = 0; k0 < kH; k0 += 4) {
      const int kk = k0 + 2 * kh;
      v2f a, b;
      a.x = Abuf[l15][kk];
      a.y = Abuf[l15][kk + 1];
      b.x = W_state[(size_t)kk * kH + n0 + l15];
      b.y = W_state[(size_t)(kk + 1) * kH + n0 + l15];
      acc = wmma4(a, b, acc);
    }
    const float bias = b_state[n0 + l15];
#pragma unroll
    for (int r = 0; r < 8; ++r)
      flat[r + 8 * kh][kH + n0 + l15] = acc[r] + bias;
  }
  __syncthreads();

  // ---- reload Abuf with stack_top tile (stack[:,0,:])
  for (int i = tid; i < 16 * 32; i += 256) {
    int r = i >> 5, c4 = i & 31;
    *(float4*)&Abuf[r][c4 * 4] =
        *(const float4*)(stack + (size_t)(m0 + r) * kD * kW + c4 * 4);
  }
  __syncthreads();

  // ---- stage A2: top_emb = stack_top @ W_top + b_top  (K = 128)
  {
    v8f acc = {};
#pragma unroll 8
    for (int k0 = 0; k0 < kH; k0 += 4) {
      const int kk = k0 + 2 * kh;
      v2f a, b;
      a.x = Abuf[l15][kk];
      a.y = Abuf[l15][kk + 1];
      b.x = W_top[(size_t)kk * kH + n0 + l15];
      b.y = W_top[(size_t)(kk + 1) * kH + n0 + l15];
      acc = wmma4(a, b, acc);
    }
    const float bias = b_top[n0 + l15];
#pragma unroll
    for (int r = 0; r < 8; ++r)
      flat[r + 8 * kh][2 * kH + n0 + l15] = acc[r] + bias;
  }
  __syncthreads();

  // ---- stage B: logits_buf and logits_state from flat (K = 384), dual accumulators
  {
    v8f ab = {}, av = {};
#pragma unroll 4
    for (int k0 = 0; k0 < kKF; k0 += 4) {
      const int kk = k0 + 2 * kh;
      v2f a;
      a.x = flat[l15][kk];
      a.y = flat[l15][kk + 1];
      const size_t base = (size_t)kk * kH + n0 + l15;
      v2f wb, ws;
      wb.x = W_buf[base];
      wb.y = W_buf[base + kH];
      ws.x = W_so[base];
      ws.y = W_so[base + kH];
      ab = wmma4(a, wb, ab);
      av = wmma4(a, ws, av);
    }
    const float bb = b_buf[n0 + l15];
    const float bs = b_so[n0 + l15];
#pragma unroll
    for (int r = 0; r < 8; ++r) {
      const size_t row = (size_t)m0 + r + 8 * kh;
      out_buf[row * kH + n0 + l15]   = ab[r] + bb;
      out_state[row * kH + n0 + l15] = tanhf(av[r] + bs);
    }
  }

  // ---- stage C: logits_mem (384 x 3 head) + softmax -> action_probs
  if (tid < 48) {
    const int r = tid / 3, j = tid % 3;
    float acc = b_mem[j];
    for (int k = 0; k < kKF; ++k) acc += flat[r][k] * W_mem[k * 3 + j];
    lm[r][j] = acc;
  }
  __syncthreads();
  if (tid < 16) {
    const float a0 = lm[tid][0], a1 = lm[tid][1], a2 = lm[tid][2];
    const float mx = fmaxf(a0, fmaxf(a1, a2));
    const float e0 = __expf(a0 - mx), e1 = __expf(a1 - mx), e2 = __expf(a2 - mx);
    const float inv = 1.0f / (e0 + e1 + e2);
    const size_t row = (size_t)m0 + tid;
    out_probs[row * 3 + 0] = e0 * inv;
    out_probs[row * 3 + 1] = e1 * inv;
    out_probs[row * 3 + 2] = e2 * inv;
  }
}

// One block per batch element: stage 32KB stack slab in LDS, blend shifted rows,
// write once. Exactly 1x read + 1x write of the 256MB tensor (HBM roofline).
__global__ __launch_bounds__(256)
void srnn_stack_kernel(const float* __restrict__ stack,
                       const float* __restrict__ probs,
                       float* __restrict__ out) {
  __shared__ float4 s[kD * kW / 4];  // 2048 float4 = 32 KB
  const int b = blockIdx.x;
  const float4* src = (const float4*)(stack + (size_t)b * kD * kW);
  float4* dst = (float4*)(out + (size_t)b * kD * kW);
#pragma unroll
  for (int i = threadIdx.x; i < kD * kW / 4; i += 256) s[i] = src[i];
  const float p0 = probs[(size_t)b * 3 + 0];
  const float p1 = probs[(size_t)b * 3 + 1];
  const float p2 = probs[(size_t)b * 3 + 2];
  __syncthreads();
#pragma unroll
  for (int i = threadIdx.x; i < kD * kW / 4; i += 256) {
    const int d = i >> 5;  // 32 float4 per row of 128
    float4 cur = s[i];
    float4 up = (d == 0) ? cur : s[i - 32];                 // "pushed"
    float4 dn = (d == kD - 1) ? make_float4(0.f, 0.f, 0.f, 0.f)
                              : s[i + 32];                  // "popped"
    float4 o;
    o.x = p0 * up.x + p1 * dn.x + p2 * cur.x;
    o.y = p0 * up.y + p1 * dn.y + p2 * cur.y;
    o.z = p0 * up.z + p1 * dn.z + p2 * cur.z;
    o.w = p0 * up.w + p1 * dn.w + p2 * cur.w;
    dst[i] = o;
  }
}

extern "C" void kernel_launch(void* const* d_in, const int* in_sizes, int n_in,
                              void* d_out, int out_size, void* d_ws, size_t ws_size,
                              hipStream_t stream) {
  (void)in_sizes; (void)n_in; (void)d_ws; (void)ws_size; (void)out_size;
  const float* stack      = (const float*)d_in[0];
  const float* state_prev = (const float*)d_in[1];
  const float* x_emb      = (const float*)d_in[2];
  const float* W_state    = (const float*)d_in[3];
  const float* b_state    = (const float*)d_in[4];
  const float* W_top      = (const float*)d_in[5];
  const float* b_top      = (const float*)d_in[6];
  const float* W_mem      = (const float*)d_in[7];
  const float* b_mem      = (const float*)d_in[8];
  const float* W_buf      = (const float*)d_in[9];
  const float* b_buf      = (const float*)d_in[10];
  const float* W_so       = (const float*)d_in[11];
  const float* b_so       = (const float*)d_in[12];

  float* out = (float*)d_out;
  const size_t off_state = (size_t)kB * kD * kW;           // after stack_new
  const size_t off_buf   = off_state + (size_t)kB * kH;    // after next_state
  const size_t off_probs = off_buf + (size_t)kB * kH;      // after logits_buf

  srnn_gemm_kernel<<<kB / 16, 256, 0, stream>>>(
      stack, state_prev, x_emb, W_state, b_state, W_top, b_top, W_mem, b_mem,
      W_buf, b_buf, W_so, b_so, out + off_state, out + off_buf, out + off_probs);

  srnn_stack_kernel<<<kB, 256, 0, stream>>>(stack, out + off_probs, out);
}